// GCNLayer_1683627180107
// MI455X (gfx1250) — compile-verified
//
#include <hip/hip_runtime.h>
#include <hip/hip_bf16.h>

// ---------------------------------------------------------------------------
// GCN layer for MI455X (gfx1250):
//   agg = segment_sum(edge_w * features[edge_src], edge_dst)   (atomic f32, L2)
//   out = relu(agg @ W + b)                                    (V_WMMA_F32_16X16X4_F32)
//
// Both phases are memory-bound (GEMM: 0.82 GFLOP over ~51 MB), so fp32 WMMA
// is the right choice; the focus is clean data movement around the matrix op.
// ---------------------------------------------------------------------------

typedef __attribute__((ext_vector_type(2))) float v2f;
typedef __attribute__((ext_vector_type(8))) float v8f;

#define D_FEAT 64
#define D_OUT  64

// ---------------------------------------------------------------------------
// Kernel 1: zero the aggregation workspace (float4-wide stores).
// ---------------------------------------------------------------------------
__global__ void __launch_bounds__(256) gcn_zero_kernel(float4* __restrict__ p, long n4) {
    long i = (long)blockIdx.x * blockDim.x + threadIdx.x;
    if (i < n4) {
        p[i] = make_float4(0.f, 0.f, 0.f, 0.f);
    }
}

// ---------------------------------------------------------------------------
// Kernel 2: edge scatter. 16 threads per edge, each handles a float4 chunk of
// the 64-wide feature row: gather features[src], scale by w, atomic-add into
// agg[dst]. Both tables are L2-resident (25.6 MB each vs 192 MB L2), so the
// random gathers and f32 atomics resolve in L2.
// ---------------------------------------------------------------------------
__global__ void __launch_bounds__(256) gcn_scatter_kernel(
    const float* __restrict__ features,
    const int* __restrict__ edge_src,
    const int* __restrict__ edge_dst,
    const float* __restrict__ edge_w,
    float* __restrict__ agg,
    long n_work)   // n_edges * 16
{
    long idx = (long)blockIdx.x * blockDim.x + threadIdx.x;
    if (idx >= n_work) return;
    long e = idx >> 4;          // edge id
    int  c = (int)(idx & 15);   // which float4 chunk (0..15)

    int src = edge_src[e];
    int dst = edge_dst[e];
    float w = edge_w[e];

    const float4* frow = (const float4*)(features + (size_t)src * D_FEAT);
    float4 f = frow[c];

    float* drow = agg + (size_t)dst * D_FEAT + (size_t)c * 4;
    atomicAdd(drow + 0, w * f.x);
    atomicAdd(drow + 1, w * f.y);
    atomicAdd(drow + 2, w * f.z);
    atomicAdd(drow + 3, w * f.w);
}

// ---------------------------------------------------------------------------
// Kernel 3: out = relu(agg @ W + b) via V_WMMA_F32_16X16X4_F32.
//
// Block = 256 threads = 8 waves; wave w owns rows [m0, m0+16). Each wave
// holds four 16x16 f32 accumulators covering N = 0..63 and iterates K in
// steps of 4 (16 WMMA k-steps).
//
// W is staged in LDS *pre-swizzled into B-fragment order*: the pair
// (W[k][n], W[k+1][n]) for even k is stored contiguously, so each lane's
// B fragment is one conflict-free ds_load_b64 into an aligned VGPR pair
// (no cross-register v_mov packing, fewer dscnt stalls).
//
// Fragment layouts (ISA 7.12.2, 32-bit):
//   A 16x4 : lane L(<16): VGPR0=A[M=L][kk+0], VGPR1=A[M=L][kk+1]
//            lanes 16-31: K offsets +2,+3  -> ka = kk + 2*(lane>>4)
//   B 4x16 : VGPR0 = B[ka][n0+lane%16], VGPR1 = B[ka+1][n0+lane%16]
//   C/D    : VGPR j = out[m0 + j + 8*(lane>>4)][n0 + lane%16]
// N_NODES % 16 == 0, so active waves always own full 16-row tiles
// (EXEC all-1s across the WMMA, as the ISA requires).
// ---------------------------------------------------------------------------
__global__ void __launch_bounds__(256) gcn_gemm_kernel(
    const float* __restrict__ agg,
    const float* __restrict__ W,    // [64 x 64] row-major (K x N)
    const float* __restrict__ bias, // [64]
    float* __restrict__ out,        // [n_nodes x 64]
    int n_nodes)
{
    __shared__ float sWz[D_FEAT * D_OUT];  // B-fragment-swizzled W
    __shared__ float sB[D_OUT];

    // Swizzle on the way in: sWz[((k>>1)*64 + n)*2 + (k&1)] = W[k][n]
    for (int i = threadIdx.x; i < D_FEAT * D_OUT; i += 256) {
        int k = i >> 6;
        int n = i & 63;
        sWz[(((k >> 1) << 6) + n) * 2 + (k & 1)] = W[i];
    }
    if (threadIdx.x < D_OUT) sB[threadIdx.x] = bias[threadIdx.x];
    __syncthreads();

    const int wave = threadIdx.x >> 5;
    const int lane = threadIdx.x & 31;
    const int half = lane >> 4;   // 0: lanes 0-15, 1: lanes 16-31
    const int l16  = lane & 15;

    const int m0 = blockIdx.x * 128 + wave * 16;
    if (m0 >= n_nodes) return;    // wave-uniform exit; no further barriers

    v8f acc0 = {}, acc1 = {}, acc2 = {}, acc3 = {};

    const float* arow = agg + (size_t)(m0 + l16) * D_FEAT;
    const v2f*   sWp  = (const v2f*)sWz;   // float2-indexed fragment table

#pragma unroll
    for (int kk = 0; kk < D_FEAT; kk += 4) {
        const int kp = (kk >> 1) + half;   // (ka>>1) with ka = kk + 2*half

        v2f a;
        a.x = arow[kk + 2 * half + 0];
        a.y = arow[kk + 2 * half + 1];

        v2f b0 = sWp[(kp << 6) +  0 + l16];
        v2f b1 = sWp[(kp << 6) + 16 + l16];
        v2f b2 = sWp[(kp << 6) + 32 + l16];
        v2f b3 = sWp[(kp << 6) + 48 + l16];

        // 8 args: (neg_a, A, neg_b, B, c_mod, C, reuse_a, reuse_b)
        acc0 = __builtin_amdgcn_wmma_f32_16x16x4_f32(false, a, false, b0, (short)0, acc0, false, false);
        acc1 = __builtin_amdgcn_wmma_f32_16x16x4_f32(false, a, false, b1, (short)0, acc1, false, false);
        acc2 = __builtin_amdgcn_wmma_f32_16x16x4_f32(false, a, false, b2, (short)0, acc2, false, false);
        acc3 = __builtin_amdgcn_wmma_f32_16x16x4_f32(false, a, false, b3, (short)0, acc3, false, false);
    }

    // Epilogue: bias + relu + store. D-matrix layout: VGPR j holds
    // row m0 + j + 8*half, column n-tile + l16.
#pragma unroll
    for (int j = 0; j < 8; ++j) {
        const int m = m0 + j + 8 * half;
        float* orow = out + (size_t)m * D_OUT;

        float v0 = acc0[j] + sB[ 0 + l16];
        float v1 = acc1[j] + sB[16 + l16];
        float v2 = acc2[j] + sB[32 + l16];
        float v3 = acc3[j] + sB[48 + l16];
        orow[ 0 + l16] = fmaxf(v0, 0.f);
        orow[16 + l16] = fmaxf(v1, 0.f);
        orow[32 + l16] = fmaxf(v2, 0.f);
        orow[48 + l16] = fmaxf(v3, 0.f);
    }
}

// ---------------------------------------------------------------------------
// Launch: zero agg -> edge scatter -> WMMA GEMM (all on `stream`, in order).
// ---------------------------------------------------------------------------
extern "C" void kernel_launch(void* const* d_in, const int* in_sizes, int n_in,
                              void* d_out, int out_size, void* d_ws, size_t ws_size,
                              hipStream_t stream) {
    const float* features = (const float*)d_in[0];
    const int*   edge_src = (const int*)d_in[1];
    const int*   edge_dst = (const int*)d_in[2];
    const float* edge_w   = (const float*)d_in[3];
    const float* W        = (const float*)d_in[4];
    const float* bias     = (const float*)d_in[5];
    float*       out      = (float*)d_out;

    const int n_nodes = in_sizes[0] / D_FEAT;
    const int n_edges = in_sizes[1];

    float* agg = (float*)d_ws;   // n_nodes * 64 floats (25.6 MB)

    // 1) zero aggregation buffer
    {
        long n4 = (long)n_nodes * D_FEAT / 4;
        long nblk = (n4 + 255) / 256;
        gcn_zero_kernel<<<(dim3)(unsigned)nblk, 256, 0, stream>>>((float4*)agg, n4);
    }

    // 2) scatter-add edges
    {
        long n_work = (long)n_edges * 16;
        long nblk = (n_work + 255) / 256;
        gcn_scatter_kernel<<<(dim3)(unsigned)nblk, 256, 0, stream>>>(
            features, edge_src, edge_dst, edge_w, agg, n_work);
    }

    // 3) dense transform with WMMA
    {
        int nblk = (n_nodes + 127) / 128;
        gcn_gemm_kernel<<<nblk, 256, 0, stream>>>(agg, W, bias, out, n_nodes);
    }
}